// ChampferLoss_86844238725722
// MI455X (gfx1250) — compile-verified
//
#include <hip/hip_runtime.h>

// Chamfer loss, restructured as:
//   d2c[n,m] = max(a2[n] + b2[m] - 2*a.b, 0)      (masked -> 1e16 = BIG^2)
//   out[b]   = sum_n ow*min_m d2c  +  sum_m tw*min_n d2c
// Cross term a.b computed with V_WMMA_F32_16X16X4_F32 (D=16 -> 4 chained WMMAs
// per 16x16 tile), with the -2 folded into the A fragment. Distance matrix is
// never materialized: row mins reduce in registers, column mins via u32
// atomic-min on float bit patterns (valid: all values are non-negative).

typedef float v2f __attribute__((ext_vector_type(2)));
typedef float v4f __attribute__((ext_vector_type(4)));
typedef float v8f __attribute__((ext_vector_type(8)));

constexpr int Bn = 32, Nn = 2048, Mn = 2048, Dn = 16;
#define BIG2 1e16f   // (1e8)^2 : masked value in dist^2 domain

// ---- init: seed column-min workspace with BIG2 bit pattern; zero out[] ----
__global__ void init_ws(unsigned* __restrict__ colws, float* __restrict__ out) {
  int i = blockIdx.x * blockDim.x + threadIdx.x;
  if (i < Bn * Mn) colws[i] = __float_as_uint(BIG2);
  if (i < Bn)      out[i]   = 0.0f;
}

// ---- main tile kernel: one wave32 per (batch, 16-row panel) ----
__global__ __launch_bounds__(32)
void chamfer_tile(const float* __restrict__ ow,   // [B,N]
                  const float* __restrict__ outp, // [B,N,D]
                  const float* __restrict__ tw,   // [B,M]
                  const float* __restrict__ tgt,  // [B,M,D]
                  unsigned*    __restrict__ colws,// [B,M] u32 min(d2c) over n (ow-masked)
                  float*       __restrict__ out)  // [B]
{
  const int b    = blockIdx.y;
  const int n0   = blockIdx.x * 16;
  const int lane = threadIdx.x;       // 0..31 (wave32)
  const int lr   = lane & 15;         // row/col within tile
  const int kh   = (lane >> 4) * 2;   // K-pair selector for 32-bit A/B frags

  const float* Aptr  = outp + (size_t)(b * Nn + n0) * Dn;
  const float* Bbase = tgt  + (size_t)b * Mn * Dn;

  // A fragments (16x4 f32, per ISA layout: lane lr = row, VGPR pair = K kh,kh+1),
  // pre-scaled by -2 so WMMA accumulates -2*a.b directly.
  v2f afrag[4];
#pragma unroll
  for (int j = 0; j < 4; ++j) {
    v2f t = *(const v2f*)(Aptr + lr * Dn + 4 * j + kh);
    afrag[j].x = -2.0f * t.x;
    afrag[j].y = -2.0f * t.y;
  }

  // Per-lane state for the 8 rows this lane-half owns in the C/D layout:
  // lanes 0-15 -> rows n0+r, lanes 16-31 -> rows n0+8+r  (r = VGPR index 0..7)
  float a2r[8], oww[8], rmin[8];
#pragma unroll
  for (int r = 0; r < 8; ++r) {
    int n = r + 8 * (lane >> 4);
    const float* rp = Aptr + n * Dn;
    float s = 0.0f;
#pragma unroll
    for (int d = 0; d < Dn; d += 4) {
      v4f q = *(const v4f*)(rp + d);
      s += q.x * q.x + q.y * q.y + q.z * q.z + q.w * q.w;
    }
    a2r[r]  = s;
    oww[r]  = ow[(size_t)b * Nn + n0 + n];
    rmin[r] = BIG2;
  }

  for (int m0 = 0; m0 < Mn; m0 += 16) {
    // B fragments (4x16 f32, B[k][c] = targets[m0+c][k]) — same lane pattern as A.
    const float* Bp = Bbase + (size_t)m0 * Dn;
    v2f bfrag[4];
    float b2p = 0.0f;
#pragma unroll
    for (int j = 0; j < 4; ++j) {
      bfrag[j] = *(const v2f*)(Bp + lr * Dn + 4 * j + kh);
      b2p += bfrag[j].x * bfrag[j].x + bfrag[j].y * bfrag[j].y;
    }
    // lanes l and l+16 jointly hold all 16 D-elements of column m0+lr
    float b2  = b2p + __shfl_xor(b2p, 16, 32);
    float twc = tw[(size_t)b * Mn + m0 + lr];

    // acc = -2 * A.B  via 4 chained f32 WMMAs (K = 0..3, 4..7, 8..11, 12..15)
    v8f acc = {};
    acc = __builtin_amdgcn_wmma_f32_16x16x4_f32(false, afrag[0], false, bfrag[0], (short)0, acc, false, false);
    acc = __builtin_amdgcn_wmma_f32_16x16x4_f32(false, afrag[1], false, bfrag[1], (short)0, acc, false, false);
    acc = __builtin_amdgcn_wmma_f32_16x16x4_f32(false, afrag[2], false, bfrag[2], (short)0, acc, false, false);
    acc = __builtin_amdgcn_wmma_f32_16x16x4_f32(false, afrag[3], false, bfrag[3], (short)0, acc, false, false);

    const bool tmask = (twc != 0.0f);
    float cmin = BIG2;
#pragma unroll
    for (int r = 0; r < 8; ++r) {
      float v = fmaxf(a2r[r] + b2 + acc[r], 0.0f);       // d2 clamped
      rmin[r] = fminf(rmin[r], tmask ? v : BIG2);        // row min: tw mask
      cmin    = fminf(cmin, (oww[r] != 0.0f) ? v : BIG2);// col min: ow mask
    }
    // combine the two lane-halves (same column), then one atomic per column
    cmin = fminf(cmin, __shfl_xor(cmin, 16, 32));
    if (lane < 16)
      atomicMin(colws + (size_t)b * Mn + m0 + lr, __float_as_uint(cmin));
  }

  // Row mins: tree-min across the 16 lanes of each half, then weighted sum.
#pragma unroll
  for (int r = 0; r < 8; ++r) {
    float v = rmin[r];
    v = fminf(v, __shfl_xor(v, 1, 32));
    v = fminf(v, __shfl_xor(v, 2, 32));
    v = fminf(v, __shfl_xor(v, 4, 32));
    v = fminf(v, __shfl_xor(v, 8, 32));
    rmin[r] = v;
  }
  if (lr == 0) {  // lane 0 (rows n0..n0+7) and lane 16 (rows n0+8..n0+15)
    float s = 0.0f;
#pragma unroll
    for (int r = 0; r < 8; ++r) s += oww[r] * rmin[r];
    atomicAdd(out + b, s);
  }
}

// ---- finalize: out[b] += sum_m tw[b,m] * colmin[b,m] ----
__global__ void chamfer_finalize(const float* __restrict__ tw,
                                 const unsigned* __restrict__ colws,
                                 float* __restrict__ out) {
  __shared__ float red[256];
  const int b = blockIdx.x;
  float s = 0.0f;
  for (int m = threadIdx.x; m < Mn; m += blockDim.x)
    s += tw[(size_t)b * Mn + m] * __uint_as_float(colws[(size_t)b * Mn + m]);
  red[threadIdx.x] = s;
  __syncthreads();
  for (int off = 128; off > 0; off >>= 1) {
    if ((int)threadIdx.x < off) red[threadIdx.x] += red[threadIdx.x + off];
    __syncthreads();
  }
  if (threadIdx.x == 0) atomicAdd(out + b, red[0]);
}

extern "C" void kernel_launch(void* const* d_in, const int* in_sizes, int n_in,
                              void* d_out, int out_size, void* d_ws, size_t ws_size,
                              hipStream_t stream) {
  const float* ow   = (const float*)d_in[0];  // o_weights [B,N]
  const float* outp = (const float*)d_in[1];  // outputs   [B,N,D]
  const float* tw   = (const float*)d_in[2];  // t_weights [B,M]
  const float* tgt  = (const float*)d_in[3];  // targets   [B,M,D]
  float* out        = (float*)d_out;          // [B]
  unsigned* colws   = (unsigned*)d_ws;        // needs B*M*4 = 256 KB

  init_ws<<<(Bn * Mn + 255) / 256, 256, 0, stream>>>(colws, out);

  dim3 grid(Nn / 16, Bn);
  chamfer_tile<<<grid, 32, 0, stream>>>(ow, outp, tw, tgt, colws, out);

  chamfer_finalize<<<Bn, 256, 0, stream>>>(tw, colws, out);
}